// OpSpNodeMessagePassing_42666205119405
// MI455X (gfx1250) — compile-verified
//
#include <hip/hip_runtime.h>
#include <stdint.h>

// Problem constants (fixed by the reference).
#define KN   8192      // nodes per batch
#define KD   64        // feature dim
#define TILE 256       // nnz staged per block iteration
#define NTHREADS 256   // 8 waves (wave32)

// ---------------------------------------------------------------------------
// Kernel 1: zero the output (B*N*D floats) with 128-bit stores.
// ---------------------------------------------------------------------------
__global__ void spnmp_zero_f4(float4* __restrict__ out, int n4) {
  int i = blockIdx.x * blockDim.x + threadIdx.x;
  if (i < n4) out[i] = make_float4(0.f, 0.f, 0.f, 0.f);
}

// ---------------------------------------------------------------------------
// CDNA5 async global -> LDS copy (ASYNCcnt-tracked, GVS addressing mode:
// addr = SGPR64 base + VGPR32 byte offset). VDST VGPR carries the LDS byte
// address (hardware adds the wave's LDS_BASE).
// ---------------------------------------------------------------------------
__device__ __forceinline__ void async_ld_b32(unsigned lds_byte_off,
                                             int      glb_byte_off,
                                             const void* sbase) {
  asm volatile("global_load_async_to_lds_b32 %0, %1, %2"
               :
               : "v"(lds_byte_off), "v"(glb_byte_off), "s"(sbase)
               : "memory");
}

// ---------------------------------------------------------------------------
// Kernel 2: COO scatter-add SpMM.
//   - Grid-stride over 256-nnz tiles; indices/vals double-buffered in LDS via
//     async copies (s_wait_asynccnt pipelining).
//   - One nnz per half-wave: 16 lanes x float4 covers D=64 -> fully coalesced
//     256B gather of X[b,col,:] and 256B of f32 atomic adds to out[b,row,:].
// ---------------------------------------------------------------------------
__global__ void __launch_bounds__(NTHREADS)
spnmp_scatter(const int*   __restrict__ Ab,
              const int*   __restrict__ Ar,
              const int*   __restrict__ Ac,
              const float* __restrict__ Av,
              const float* __restrict__ X,
              const uint8_t* __restrict__ xmask,   // (B,N) bool
              const uint8_t* __restrict__ tmask,   // (B,N) bool
              float* __restrict__ out,
              int nnz) {
  // [2 buffers][4 arrays][TILE] ints = 8 KB
  __shared__ int smem[2 * 4 * TILE];
  const unsigned lds0 = (unsigned)(uintptr_t)(&smem[0]);  // low 32 bits = LDS offset

  const int tid = threadIdx.x;
  const int g   = tid >> 4;   // half-wave group id 0..15 -> which nnz
  const int l   = tid & 15;   // lane in group -> float4 chunk of D

  const int tiles = (nnz + TILE - 1) / TILE;

  // Stage one tile's 4 arrays into LDS buffer `buf`. Every wave issues exactly
  // 4 async instructions (addresses clamped so ASYNCcnt bookkeeping is exact).
  auto stage = [&](int buf, int tile) {
    int e = tile * TILE + tid;
    if (e > nnz - 1) e = nnz - 1;          // clamp (dup of last elem, unused)
    const unsigned lb = lds0 + (unsigned)(buf * (4 * TILE) + tid) * 4u;
    const int go = e * 4;                  // byte offset into int/f32 arrays
    async_ld_b32(lb + 0u * (TILE * 4u), go, Ab);
    async_ld_b32(lb + 1u * (TILE * 4u), go, Ar);
    async_ld_b32(lb + 2u * (TILE * 4u), go, Ac);
    async_ld_b32(lb + 3u * (TILE * 4u), go, Av);
  };

  int tile = blockIdx.x;
  int buf  = 0;
  if (tile < tiles) stage(buf, tile);      // prologue

  for (; tile < tiles; tile += gridDim.x) {
    const int  next     = tile + gridDim.x;
    const bool has_next = next < tiles;

    if (has_next) stage(buf ^ 1, next);    // overlap next tile's copy

    // Async loads retire in order: waiting <=4 outstanding guarantees the
    // current tile's 4 copies (issued first) have landed in LDS.
    if (has_next) {
      asm volatile("s_wait_asynccnt 0x4" ::: "memory");
    } else {
      asm volatile("s_wait_asynccnt 0x0" ::: "memory");
    }
    __syncthreads();                       // LDS tile visible to all 8 waves

    const int ibase = buf * (4 * TILE);
    int limit = nnz - tile * TILE;
    if (limit > TILE) limit = TILE;

    #pragma unroll 4
    for (int k = 0; k < TILE / 16; ++k) {
      const int el = k * 16 + g;           // nnz slot for this half-wave
      if (el < limit) {
        const int   b = smem[ibase + 0 * TILE + el];
        const int   r = smem[ibase + 1 * TILE + el];
        const int   c = smem[ibase + 2 * TILE + el];
        const float v = __int_as_float(smem[ibase + 3 * TILE + el]);
        const int bn_c = b * KN + c;
        const int bn_r = b * KN + r;
        // X_mask gates the gathered row; tarX_mask gates the target row
        // (out was zero-initialized, so skipping == masking).
        if (xmask[bn_c] && tmask[bn_r]) {
          const float4 xv =
              reinterpret_cast<const float4*>(X + (size_t)bn_c * KD)[l];
          float* op = out + (size_t)bn_r * KD + (l << 2);
          unsafeAtomicAdd(op + 0, v * xv.x);   // global_atomic_add_f32
          unsafeAtomicAdd(op + 1, v * xv.y);
          unsafeAtomicAdd(op + 2, v * xv.z);
          unsafeAtomicAdd(op + 3, v * xv.w);
        }
      }
    }
    __syncthreads();                       // done reading buf before re-stage
    buf ^= 1;
  }
}

// ---------------------------------------------------------------------------
// Launcher. Inputs (reference order): A_batch, A_row, A_col, A_vals, X,
// X_mask, tarX_mask. Output: (B,N,D) float32.
// ---------------------------------------------------------------------------
extern "C" void kernel_launch(void* const* d_in, const int* in_sizes, int n_in,
                              void* d_out, int out_size, void* d_ws, size_t ws_size,
                              hipStream_t stream) {
  const int*     Ab = (const int*)d_in[0];
  const int*     Ar = (const int*)d_in[1];
  const int*     Ac = (const int*)d_in[2];
  const float*   Av = (const float*)d_in[3];
  const float*   X  = (const float*)d_in[4];
  const uint8_t* xm = (const uint8_t*)d_in[5];
  const uint8_t* tm = (const uint8_t*)d_in[6];
  float* out = (float*)d_out;

  const int nnz = in_sizes[0];

  // 1) zero-fill output (also applies tarX_mask implicitly).
  const int n4 = out_size / 4;
  spnmp_zero_f4<<<(n4 + NTHREADS - 1) / NTHREADS, NTHREADS, 0, stream>>>(
      (float4*)out, n4);

  // 2) scatter-add. Persistent-ish grid so each block pipelines several tiles.
  const int tiles  = (nnz + TILE - 1) / TILE;
  int blocks = tiles < 1024 ? tiles : 1024;
  if (blocks < 1) blocks = 1;
  spnmp_scatter<<<blocks, NTHREADS, 0, stream>>>(Ab, Ar, Ac, Av, X, xm, tm,
                                                 out, nnz);
}